// NewModel_29291676959125
// MI455X (gfx1250) — compile-verified
//
#include <hip/hip_runtime.h>
#include <hip/hip_bf16.h>

typedef float v2f __attribute__((ext_vector_type(2)));
typedef float v8f __attribute__((ext_vector_type(8)));

#define WAVES   4
#define TILES   2          // token tiles per wave (two accumulators)
#define TILE_M  16
#define BK      64         // k-slab per iteration
#define AST     68         // padded LDS row stride (dwords): aligned b128 stores,
                           // conflict-free ds_load_b64 A-fragment reads
#define THREADS (WAVES * 32)

// Low 32 bits of a generic pointer to __shared__ = wave-relative LDS byte offset
// (ISA: flat->LDS uses addr[31:0]; generic AS3 pointers are aperture|offset).
__device__ __forceinline__ unsigned lds_off(const void* p) {
    return (unsigned)(unsigned long long)p;
}

__device__ __forceinline__ void async_b128_to_lds(unsigned lds, const void* gaddr) {
    asm volatile("global_load_async_to_lds_b128 %0, %1, off"
                 :: "v"(lds), "v"(gaddr) : "memory");
}

// ---------------------------------------------------------------------------
// Prep kernel: pack Wc[H][16] (= [gate(4) | expert e,o(8) | pad(4)]) into the
// WMMA-B-fragment interleaved layout: entry (kk, n) = (Wc[2kk][n], Wc[2kk+1][n]).
// Runs once per launch; result lives in L2 (256 KB) for the main kernel.
// ---------------------------------------------------------------------------
__global__ void pack_w_kernel(const float* __restrict__ gate_w,
                              const float* __restrict__ expert_w,
                              float* __restrict__ w2g, int H)
{
    int idx = blockIdx.x * blockDim.x + threadIdx.x;   // 0 .. (H/2)*16-1
    if (idx >= (H / 2) * 16) return;
    int kk = idx >> 4, n = idx & 15, k = 2 * kk;
    float x0 = 0.0f, x1 = 0.0f;
    if (n < 4) {                       // gate_w [H,4]
        x0 = gate_w[(size_t)k * 4 + n];
        x1 = gate_w[(size_t)(k + 1) * 4 + n];
    } else if (n < 12) {               // expert_w [4,H,2]
        int e = (n - 4) >> 1, o = (n - 4) & 1;
        const float* wp = expert_w + (size_t)e * H * 2 + o;
        x0 = wp[(size_t)k * 2];
        x1 = wp[(size_t)(k + 1) * 2];
    }
    w2g[(size_t)idx * 2 + 0] = x0;
    w2g[(size_t)idx * 2 + 1] = x1;
}

__device__ __forceinline__ void token_epilogue(const float* __restrict__ row,
                                               const float* __restrict__ gate_b,
                                               const float* __restrict__ expert_b,
                                               float* __restrict__ out, int token)
{
    float g[4], y[8];
    #pragma unroll
    for (int e = 0; e < 4; ++e) g[e] = row[e] + gate_b[e];
    #pragma unroll
    for (int j = 0; j < 8; ++j) y[j] = row[4 + j] + expert_b[j];
    // top-2, first index wins ties (matches jax.lax.top_k)
    int i1 = 0;
    #pragma unroll
    for (int e = 1; e < 4; ++e) if (g[e] > g[i1]) i1 = e;
    int i2 = (i1 == 0) ? 1 : 0;
    #pragma unroll
    for (int e = 0; e < 4; ++e) if (e != i1 && g[e] > g[i2]) i2 = e;
    const float d  = __expf(g[i2] - g[i1]);       // softmax over top-2 pair
    const float w1 = 1.0f / (1.0f + d);
    const float w2 = d * w1;
    out[(size_t)token * 2 + 0] = w1 * y[i1 * 2 + 0] + w2 * y[i2 * 2 + 0];
    out[(size_t)token * 2 + 1] = w1 * y[i1 * 2 + 1] + w2 * y[i2 * 2 + 1];
}

// ---------------------------------------------------------------------------
// Main kernel: D[32 tok x 16 col] = hidden_tile * Wc via V_WMMA_F32_16X16X4_F32.
// Double-buffered async global->LDS staging (ASYNCcnt), 2 accumulators/wave.
// ---------------------------------------------------------------------------
__global__ __launch_bounds__(THREADS)
void moe_wmma_kernel(const float* __restrict__ hidden,
                     const float* __restrict__ w2g,
                     const float* __restrict__ gate_b,
                     const float* __restrict__ expert_b,
                     float* __restrict__ out,
                     int N, int H)
{
    __shared__ float W2[2][(BK / 2) * 16 * 2];                  // [buf][1024] = 8 KB
    __shared__ float At[WAVES][2][TILES][TILE_M * AST];         // ~68 KB

    const int tid    = threadIdx.x;
    const int wave   = tid >> 5;
    const int lane   = tid & 31;
    const int laneLo = lane & 15;
    const int laneHi = lane >> 4;

    const int m0 = blockIdx.x * (WAVES * TILES * TILE_M) + wave * (TILES * TILE_M);

    v8f acc0 = {};
    v8f acc1 = {};

    // Issue one k-slab's async copies (18 per wave: 16 A-tile + 2 W-slab).
    auto issue_slab = [&](int k0, int buf) {
        const int c = laneLo * 4;
        #pragma unroll
        for (int t = 0; t < TILES; ++t) {
            #pragma unroll
            for (int i = 0; i < 8; ++i) {
                int r = 2 * i + laneHi;     // 2 rows x 256B per b128 instruction
                const float* gp = hidden + (size_t)(m0 + t * TILE_M + r) * H + k0 + c;
                async_b128_to_lds(lds_off(&At[wave][buf][t][r * AST + c]), gp);
            }
        }
        // W slab: 1024 contiguous floats, 128 threads x 2 float4 — fully coalesced
        const float* gp0 = w2g + (size_t)k0 * 16 + tid * 4;
        unsigned l0 = lds_off(&W2[buf][tid * 4]);
        async_b128_to_lds(l0, gp0);
        async_b128_to_lds(l0 + 512 * 4, gp0 + 512);
    };

    const int nslab = H / BK;
    issue_slab(0, 0);

    for (int kb = 0; kb < nslab; ++kb) {
        const int buf = kb & 1;
        if (kb + 1 < nslab) {
            issue_slab((kb + 1) * BK, buf ^ 1);
            // next slab's 18 asyncs may remain outstanding; this slab must be done
            asm volatile("s_wait_asynccnt 18" ::: "memory");
        } else {
            asm volatile("s_wait_asynccnt 0" ::: "memory");
        }
        __syncthreads();

        #pragma unroll
        for (int s = 0; s < BK / 4; ++s) {
            // B frag: lane n<16 -> (K=0,1 col n); lane n>=16 -> (K=2,3 col n)
            v2f b  = *(const v2f*)(&W2[buf][((2 * s + laneHi) * 16 + laneLo) * 2]);
            // A frag: lane l<16 -> (row l, k..k+1); lane l>=16 -> (row l-16, k+2..k+3)
            v2f a0 = *(const v2f*)(&At[wave][buf][0][laneLo * AST + 4 * s + 2 * laneHi]);
            v2f a1 = *(const v2f*)(&At[wave][buf][1][laneLo * AST + 4 * s + 2 * laneHi]);
            acc0 = __builtin_amdgcn_wmma_f32_16x16x4_f32(false, a0, false, b,
                                                         (short)0, acc0, false, false);
            acc1 = __builtin_amdgcn_wmma_f32_16x16x4_f32(false, a1, false, b,
                                                         (short)0, acc1, false, false);
        }
        __syncthreads();   // all reads of `buf` done before it is overwritten
    }

    // ---- transpose D tiles through LDS so one lane owns one token ----
    float* ep0 = &At[wave][0][0][0];
    float* ep1 = &At[wave][0][1][0];
    #pragma unroll
    for (int r = 0; r < 8; ++r) {
        // VGPR r, lanes 0-15: M=r; lanes 16-31: M=r+8 (N = laneLo)
        ep0[(r + 8 * laneHi) * 16 + laneLo] = acc0[r];
        ep1[(r + 8 * laneHi) * 16 + laneLo] = acc1[r];
    }
    __syncthreads();

    if (lane < 16) {
        int t0 = m0 + lane;
        int t1 = m0 + TILE_M + lane;
        if (t0 < N) token_epilogue(ep0 + lane * 16, gate_b, expert_b, out, t0);
        if (t1 < N) token_epilogue(ep1 + lane * 16, gate_b, expert_b, out, t1);
    }
}

extern "C" void kernel_launch(void* const* d_in, const int* in_sizes, int n_in,
                              void* d_out, int out_size, void* d_ws, size_t ws_size,
                              hipStream_t stream) {
    const float* hidden   = (const float*)d_in[0];
    const float* gate_w   = (const float*)d_in[1];
    const float* gate_b   = (const float*)d_in[2];
    const float* expert_w = (const float*)d_in[3];
    const float* expert_b = (const float*)d_in[4];
    float* out = (float*)d_out;
    float* w2g = (float*)d_ws;            // packed Wc, H*16 floats (256 KB for H=4096)

    const int H = in_sizes[1] / 4;        // gate_w is [H, 4]
    const int N = in_sizes[0] / H;        // hidden is [N, H]

    // 1) pack weights into WMMA-B-friendly layout (one-shot, tiny)
    const int wtotal = (H / 2) * 16;
    pack_w_kernel<<<(wtotal + 255) / 256, 256, 0, stream>>>(gate_w, expert_w, w2g, H);

    // 2) fused MoE GEMM + top-2 softmax combine
    const int tokens_per_wg = WAVES * TILES * TILE_M;   // 128
    const int grid = (N + tokens_per_wg - 1) / tokens_per_wg;
    moe_wmma_kernel<<<grid, THREADS, 0, stream>>>(hidden, w2g, gate_b, expert_b, out, N, H);
}